// HierarchicalDiscoExtractor_26869315403891
// MI455X (gfx1250) — compile-verified
//
#include <hip/hip_runtime.h>

typedef __attribute__((ext_vector_type(2))) float v2f;
typedef __attribute__((ext_vector_type(4))) float v4f;
typedef __attribute__((ext_vector_type(8))) float v8f;

#define H_LAT 91
#define W_LON 180
#define HW (H_LAT * W_LON)

// ---------------------------------------------------------------------------
// CSR row pointers from sorted io_idx (lower_bound per output latitude)
// ---------------------------------------------------------------------------
__global__ void rowstart_kernel(const int* __restrict__ io_idx,
                                int* __restrict__ rowptr, int N) {
  int h = threadIdx.x;
  if (h > H_LAT) return;                    // rowptr[0..91]
  int lo = 0, hi = N;
  while (lo < hi) {
    int mid = (lo + hi) >> 1;
    if (io_idx[mid] < h) lo = mid + 1; else hi = mid;
  }
  rowptr[h] = lo;
}

// meta[n] = (d<<16) | ii  ; d = widx[n,0]
__global__ void meta_kernel(const int* __restrict__ ii_idx,
                            const int* __restrict__ widx,
                            int* __restrict__ meta, int N) {
  int n = blockIdx.x * 256 + threadIdx.x;
  if (n < N) meta[n] = (widx[(size_t)n * W_LON] << 16) | ii_idx[n];
}

// Layer-1 coef: coef1[n*16 + o] = sum_k w[o*7+k]*psi[k*N+n] (o<8), 0 else.
__global__ void coef1_kernel(const float* __restrict__ wmat,
                             const float* __restrict__ psi,
                             float* __restrict__ coef, int N) {
  int idx = blockIdx.x * 256 + threadIdx.x;
  if (idx >= N * 16) return;
  int o = idx & 15;
  int n = idx >> 4;
  float s = 0.0f;
  if (o < 8) {
#pragma unroll
    for (int k = 0; k < 7; ++k)
      s += wmat[o * 7 + k] * psi[(size_t)k * N + n];
  }
  coef[idx] = s;
}

// C=8 coef in WMMA B-operand order: coefOp[((n*2+ch)*32+lane)*2+j] =
//   a[n, c = ch*4 + 2*(lane>>4) + j, o = lane&15]  (o<8, else 0)
// -> per chunk, each lane's two K values are one b64 load.
__global__ void coefop_kernel(const float* __restrict__ wmat, // [8][8][7]
                              const float* __restrict__ psi,  // [7][N]
                              float* __restrict__ coefOp, int N) {
  int idx = blockIdx.x * 256 + threadIdx.x;
  if (idx >= N * 128) return;
  int j    = idx & 1;
  int lane = (idx >> 1) & 31;
  int ch   = (idx >> 6) & 1;
  int n    = idx >> 7;
  int o = lane & 15;
  int c = ch * 4 + 2 * (lane >> 4) + j;
  float s = 0.0f;
  if (o < 8) {
#pragma unroll
    for (int k = 0; k < 7; ++k)
      s += wmat[(o * 8 + c) * 7 + k] * psi[(size_t)k * N + n];
  }
  coefOp[idx] = s;
}

// ---------------------------------------------------------------------------
// CIN=8 DISCO layer, V_WMMA_F32_16X16X4_F32, channel-last input (B,H,W,8).
// Per nnz entry n (wave-uniform): 2 chunks of K=4 channels = 2 WMMAs,
// fed by 2 x b64 activation loads + 2 x b64 operand-ordered coef loads.
// ---------------------------------------------------------------------------
__global__ __launch_bounds__(256)
void disco_wmma8(const float* __restrict__ xt,     // (8,91,180,8) channel-last
                 const float* __restrict__ coefOp, // [N][2][32][2]
                 const float* __restrict__ bias,   // [8]
                 const int* __restrict__ rowptr,   // [92]
                 const int* __restrict__ meta,     // [N]
                 float* __restrict__ outp,         // channel-base-adjusted out
                 int out_bstride,
                 float* __restrict__ xtout) {      // next layer input or null
  const int ho  = blockIdx.y;
  const int ns  = rowptr[ho];
  const int nnz = rowptr[ho + 1] - ns;

  const int waveId = threadIdx.x >> 5;
  const int lane   = threadIdx.x & 31;
  const int tile   = blockIdx.x * 8 + waveId;   // 90 tiles per output row
  if (tile >= 90) return;

  const int wg = tile % 45;                     // longitude group (4 cols)
  const int bg = tile / 45;                     // batch group (4 batches)
  const int g  = lane >> 4;                     // K pair base 2*g
  const int r  = lane & 15;                     // M for A, N(=o) for B

  const int mb = bg * 4 + (r >> 2);             // batch of A row m=r
  const int mw = wg * 4 + (r & 3);              // longitude of A row m=r
  const int c0 = 2 * g;                         // lane's first channel

  const int voffAC = mb * (HW * 8) + c0;        // per lane, loop-invariant (floats)
  const float* __restrict__ cbase =
      coefOp + (((size_t)ns) << 7) + (lane << 1);

  v8f acc = {0.f, 0.f, 0.f, 0.f, 0.f, 0.f, 0.f, 0.f};

#pragma unroll 2
  for (int n = 0; n < nnz; ++n) {
    const int md = __builtin_amdgcn_readfirstlane(meta[ns + n]);
    const int ii = md & 0xffff;                 // scalar
    const int d  = md >> 16;                    // scalar
    int wv = mw + d;
    wv = (wv >= W_LON) ? wv - W_LON : wv;       // circular longitude
    const float* __restrict__ xp =
        (xt + ii * (W_LON * 8)) + (voffAC + (wv << 3));
    const v2f A0 = *(const v2f*)(xp);           // c = c0, c0+1
    const v2f A1 = *(const v2f*)(xp + 4);       // c = c0+4, c0+5
    const float* __restrict__ cp = cbase + (((size_t)n) << 7);
    const v2f B0 = *(const v2f*)(cp);           // chunk 0
    const v2f B1 = *(const v2f*)(cp + 64);      // chunk 1
    acc = __builtin_amdgcn_wmma_f32_16x16x4_f32(false, A0, false, B0,
                                                (short)0, acc, false, false);
    acc = __builtin_amdgcn_wmma_f32_16x16x4_f32(false, A1, false, B1,
                                                (short)0, acc, false, false);
  }

  if (r < 8) {
    const float bb = bias[r];
    // D: vgpr rr -> m = 8*g+rr; rr 0..3 -> b = bg*4+2g, w contiguous
    v4f s0 = {acc[0] + bb, acc[1] + bb, acc[2] + bb, acc[3] + bb};
    v4f s1 = {acc[4] + bb, acc[5] + bb, acc[6] + bb, acc[7] + bb};
    const int b0i = bg * 4 + 2 * g;
    size_t o0 = (size_t)b0i * out_bstride + r * HW + ho * W_LON + wg * 4;
    *(v4f*)(outp + o0) = s0;
    *(v4f*)(outp + o0 + out_bstride) = s1;
    if (xtout) {                                // channel-last copy for next layer
#pragma unroll
      for (int rr = 0; rr < 8; ++rr) {
        const int m = g * 8 + rr;
        const int b = bg * 4 + (m >> 2);
        const int w = wg * 4 + (m & 3);
        xtout[((size_t)(b * H_LAT + ho) * W_LON + w) * 8 + r] = acc[rr] + bb;
      }
    }
  }
}

// ---------------------------------------------------------------------------
// CIN=1 layer (layer 1, ~6% of FLOPs): generic chunked path; also emits the
// channel-last activation tensor consumed by layer 2.
// ---------------------------------------------------------------------------
__global__ __launch_bounds__(256)
void disco_wmma1(const float* __restrict__ xin,  // (8,1,91,180)
                 const float* __restrict__ coef, // [N][16] zero-padded
                 const float* __restrict__ bias,
                 const int* __restrict__ rowptr,
                 const int* __restrict__ meta,
                 float* __restrict__ outp, int out_bstride,
                 float* __restrict__ xtout) {    // (8,91,180,8)
  const int ho  = blockIdx.y;
  const int ns  = rowptr[ho];
  const int nnz = rowptr[ho + 1] - ns;

  const int waveId = threadIdx.x >> 5;
  const int lane   = threadIdx.x & 31;
  const int tile   = blockIdx.x * 8 + waveId;
  if (tile >= 90) return;

  const int wg = tile % 45, bg = tile / 45;
  const int g  = lane >> 4, r = lane & 15;
  const int mb = bg * 4 + (r >> 2);
  const int mw = wg * 4 + (r & 3);
  const int voffA = mb * HW;

  const int Krow = nnz;
  const int nchunk = (Krow + 3) >> 2;

  v8f acc = {0.f, 0.f, 0.f, 0.f, 0.f, 0.f, 0.f, 0.f};

  for (int cc = 0; cc < nchunk; ++cc) {
    const int t0 = cc * 4 + 2 * g;
    const int t1 = t0 + 1;
    const int t0c = (t0 < Krow) ? t0 : Krow - 1;
    const int t1c = (t1 < Krow) ? t1 : Krow - 1;
    const int md0 = meta[ns + t0c];
    const int md1 = meta[ns + t1c];
    int w0 = mw + (md0 >> 16); if (w0 >= W_LON) w0 -= W_LON;
    int w1 = mw + (md1 >> 16); if (w1 >= W_LON) w1 -= W_LON;
    const float a0 = xin[(size_t)voffA + (md0 & 0xffff) * W_LON + w0];
    const float a1 = xin[(size_t)voffA + (md1 & 0xffff) * W_LON + w1];
    float b0 = coef[(((size_t)(ns + t0c)) << 4) + r];
    float b1 = coef[(((size_t)(ns + t1c)) << 4) + r];
    if (t0 >= Krow) b0 = 0.f;                   // cndmask, no branch
    if (t1 >= Krow) b1 = 0.f;
    v2f A = {a0, a1}, B = {b0, b1};
    acc = __builtin_amdgcn_wmma_f32_16x16x4_f32(false, A, false, B,
                                                (short)0, acc, false, false);
  }

  if (r < 8) {
    const float bb = bias[r];
    v4f s0 = {acc[0] + bb, acc[1] + bb, acc[2] + bb, acc[3] + bb};
    v4f s1 = {acc[4] + bb, acc[5] + bb, acc[6] + bb, acc[7] + bb};
    const int b0i = bg * 4 + 2 * g;
    size_t o0 = (size_t)b0i * out_bstride + r * HW + ho * W_LON + wg * 4;
    *(v4f*)(outp + o0) = s0;
    *(v4f*)(outp + o0 + out_bstride) = s1;
#pragma unroll
    for (int rr = 0; rr < 8; ++rr) {
      const int m = g * 8 + rr;
      const int b = bg * 4 + (m >> 2);
      const int w = wg * 4 + (m & 3);
      xtout[((size_t)(b * H_LAT + ho) * W_LON + w) * 8 + r] = acc[rr] + bb;
    }
  }
}

// ---------------------------------------------------------------------------
extern "C" void kernel_launch(void* const* d_in, const int* in_sizes, int n_in,
                              void* d_out, int out_size, void* d_ws, size_t ws_size,
                              hipStream_t stream) {
  const float* x   = (const float*)d_in[0];
  const float* psi = (const float*)d_in[1];
  const float* w1  = (const float*)d_in[2];
  const float* b1  = (const float*)d_in[3];
  const float* w2  = (const float*)d_in[4];
  const float* b2  = (const float*)d_in[5];
  const float* w3  = (const float*)d_in[6];
  const float* b3  = (const float*)d_in[7];
  const int* io_idx = (const int*)d_in[8];
  const int* ii_idx = (const int*)d_in[9];
  const int* widx   = (const int*)d_in[10];
  const int N  = in_sizes[8];                      // nnz count
  const int Np = (N + 3) & ~3;                     // keep 16B sub-buffer align
  float* out = (float*)d_out;                      // (8, 24, 91, 180)

  // scratch layout (all 16B aligned)
  int*   rowptr = (int*)d_ws;                      // 128 ints
  int*   meta   = rowptr + 128;                    // Np ints
  float* coef1  = (float*)(meta + Np);             // N*16
  float* coefOp = coef1 + (size_t)N * 16;          // N*128 (reused L2/L3)
  float* xt1    = coefOp + (size_t)N * 128;        // 8*HW*8 channel-last act
  float* xt2    = xt1 + (size_t)8 * HW * 8;        // 8*HW*8

  const int OUT_BS = 24 * HW;                      // output batch stride
  const dim3 dgrid(12, H_LAT);                     // 12*8 waves >= 90 tiles

  rowstart_kernel<<<1, 128, 0, stream>>>(io_idx, rowptr, N);
  meta_kernel<<<(N + 255) / 256, 256, 0, stream>>>(ii_idx, widx, meta, N);

  // layer 1: x (B,1,H,W) -> out ch 0..7  (+ channel-last xt1)
  coef1_kernel<<<(N * 16 + 255) / 256, 256, 0, stream>>>(w1, psi, coef1, N);
  disco_wmma1<<<dgrid, 256, 0, stream>>>(x, coef1, b1, rowptr, meta,
                                         out, OUT_BS, xt1);
  // layer 2: xt1 -> out ch 8..15 (+ xt2)
  coefop_kernel<<<(N * 128 + 255) / 256, 256, 0, stream>>>(w2, psi, coefOp, N);
  disco_wmma8<<<dgrid, 256, 0, stream>>>(xt1, coefOp, b2, rowptr, meta,
                                         out + 8 * HW, OUT_BS, xt2);
  // layer 3: xt2 -> out ch 16..23
  coefop_kernel<<<(N * 128 + 255) / 256, 256, 0, stream>>>(w3, psi, coefOp, N);
  disco_wmma8<<<dgrid, 256, 0, stream>>>(xt2, coefOp, b3, rowptr, meta,
                                         out + 16 * HW, OUT_BS, nullptr);
}